// WDECTLayer_15942918603129
// MI455X (gfx1250) — compile-verified
//
#include <hip/hip_runtime.h>

typedef float v2f __attribute__((ext_vector_type(2)));
typedef float v8f __attribute__((ext_vector_type(8)));

#define NUM_GRAPHS 64
#define T_DIM 32
#define S_DIM 32
#define ACC_ELEMS (NUM_GRAPHS * S_DIM * T_DIM)   // 65536 floats = 256KB LDS
// SCALE * log2(e): sigmoid(SCALE*z) = 1 / (1 + 2^(-SCALE_K*z))
#define SCALE_K 721.3475204444817f

// ---------------- K0: zero the output accumulator ----------------
__global__ void wdect_zero(float* __restrict__ out, int n) {
  int i = blockIdx.x * blockDim.x + threadIdx.x;
  if (i < n) out[i] = 0.0f;
}

// ---------------- K1: nh[N,32] = (x * w) @ v via V_WMMA_F32_16X16X4_F32 ----
// A (16x4 f32): lanes 0-15 hold (M=lane, K=0..1) in v0/v1,
//               lanes 16-31 hold (M=lane-16, K=2..3). K=3 zero-padded (D=3).
// B (4x16 f32): symmetric — lanes 0-15 hold column N=lane for K=0..1,
//               lanes 16-31 hold column N=lane-16 for K=2..3.
// D (16x16 f32): vgpr r -> (M = r + 8*(lane>=16), N = lane&15).
__global__ __launch_bounds__(256)
void wdect_heights(const float* __restrict__ x,
                   const float* __restrict__ w,
                   const float* __restrict__ v,
                   float* __restrict__ nh, int N) {
  const int lane = threadIdx.x & 31;
  const int wavesPerBlock = blockDim.x >> 5;
  const int tile = blockIdx.x * wavesPerBlock + (threadIdx.x >> 5);
  const int nTiles = (N + 15) >> 4;
  if (tile >= nTiles) return;          // wave-uniform: EXEC stays all-ones

  const int mrow = lane & 15;
  const bool hiK = lane >= 16;
  int node = tile * 16 + mrow;
  if (node >= N) node = N - 1;         // clamp (only matters for tail tile)

  // Branchless A build: unconditional in-bounds loads + cndmask, no exec dance.
  const float wt = w[node];
  const float x_lo = x[node * 3 + (hiK ? 2 : 0)];  // K0 (lanes<16) / K2 (lanes>=16)
  const float x_k1 = x[node * 3 + 1];              // masked off for lanes>=16
  v2f a;
  a.x = x_lo * wt;
  a.y = hiK ? 0.0f : (x_k1 * wt);                  // K=3 pad on hi half

  const bool fullTile = (tile * 16 + 16) <= N;     // wave-uniform

  #pragma unroll
  for (int half = 0; half < 2; ++half) {           // t-columns [0,16) and [16,32)
    const int col = half * 16 + mrow;
    // Branchless B build, same masking trick.
    v2f b;
    b.x = v[(hiK ? 2 : 0) * T_DIM + col];
    b.y = hiK ? 0.0f : v[1 * T_DIM + col];

    v8f c = {};
    c = __builtin_amdgcn_wmma_f32_16x16x4_f32(
            /*neg_a=*/false, a, /*neg_b=*/false, b,
            /*c_mod=*/(short)0, c, /*reuse_a=*/false, /*reuse_b=*/false);

    const int rowBase = tile * 16 + (hiK ? 8 : 0);
    if (fullTile) {                                // uniform branch: hot path
      #pragma unroll
      for (int r = 0; r < 8; ++r)
        nh[(rowBase + r) * T_DIM + col] = c[r];
    } else {                                       // tail tile only
      #pragma unroll
      for (int r = 0; r < 8; ++r) {
        const int row = rowBase + r;
        if (row < N) nh[row * T_DIM + col] = c[r];
      }
    }
  }
}

// ---------------- K2: ECC accumulation over all simplices ----------------
// Lane == t. Full [64,32,32] accumulator in LDS (256KB, CDNA5 WGP has 320KB).
// ds_add_f32 per (s,t) contribution; lanes hit 32 consecutive banks.
__global__ __launch_bounds__(1024)
void wdect_ecc(const float* __restrict__ nh,
               const int*   __restrict__ batch_idx,
               const int*   __restrict__ edge_index,   // [2,E]
               const float* __restrict__ edge_w,
               const int*   __restrict__ face,         // [3,F]
               const float* __restrict__ face_w,
               const float* __restrict__ lin,
               float* __restrict__ out,
               int N, int E, int F) {
  extern __shared__ float acc[];       // ACC_ELEMS floats
  const int tid = threadIdx.x;
  for (int i = tid; i < ACC_ELEMS; i += blockDim.x) acc[i] = 0.0f;

  // thresholds pre-scaled, kept in registers (fully-unrolled inner loop)
  float kl[S_DIM];
  #pragma unroll
  for (int s = 0; s < S_DIM; ++s) kl[s] = SCALE_K * lin[s];

  __syncthreads();

  const int lane = tid & 31;                                   // == t
  const int wave = blockIdx.x * (blockDim.x >> 5) + (tid >> 5);
  const int waveStride = gridDim.x * (blockDim.x >> 5);
  const int M = N + E + F;

  for (int m = wave; m < M; m += waveStride) {     // wave-uniform simplex id
    float h, sgn;
    int g;
    if (m < N) {                                   // node
      h = nh[m * T_DIM + lane];
      g = batch_idx[m];
      sgn = 1.0f;
    } else if (m < N + E) {                        // edge (subtracted)
      const int e  = m - N;
      const int i0 = edge_index[e];
      const int i1 = edge_index[E + e];
      h = fmaxf(nh[i0 * T_DIM + lane], nh[i1 * T_DIM + lane]) * edge_w[e];
      g = batch_idx[i0];
      sgn = -1.0f;
    } else {                                       // face (added)
      const int f  = m - N - E;
      const int i0 = face[f];
      const int i1 = face[F + f];
      const int i2 = face[2 * F + f];
      const float h0 = nh[i0 * T_DIM + lane];
      const float h1 = nh[i1 * T_DIM + lane];
      const float h2 = nh[i2 * T_DIM + lane];
      h = fmaxf(fmaxf(h0, h1), h2) * face_w[f];
      g = batch_idx[i0];
      sgn = 1.0f;
    }

    const float kh = SCALE_K * h;
    float* dst = &acc[g * (S_DIM * T_DIM) + lane];
    #pragma unroll
    for (int s = 0; s < S_DIM; ++s) {
      // sgn*sigmoid = rcp(sgn*(1 + 2^(kh - kl))); exp2 overflow -> rcp(inf)=0,
      // underflow -> rcp(sgn)=sgn: exact saturation at both ends.
      const float e2  = __builtin_amdgcn_exp2f(kh - kl[s]);
      const float sig = __builtin_amdgcn_rcpf(__builtin_fmaf(e2, sgn, sgn));
      unsafeAtomicAdd(&dst[s * T_DIM], sig);       // ds_add_f32
    }
  }

  __syncthreads();
  for (int i = tid; i < ACC_ELEMS; i += blockDim.x)
    unsafeAtomicAdd(&out[i], acc[i]);              // global_atomic_add_f32
}

extern "C" void kernel_launch(void* const* d_in, const int* in_sizes, int n_in,
                              void* d_out, int out_size, void* d_ws, size_t ws_size,
                              hipStream_t stream) {
  const float* x     = (const float*)d_in[0];
  const float* nw    = (const float*)d_in[1];
  const float* ew    = (const float*)d_in[2];
  const float* fw    = (const float*)d_in[3];
  const int*   batch = (const int*)  d_in[4];
  const int*   eidx  = (const int*)  d_in[5];
  const int*   fidx  = (const int*)  d_in[6];
  const float* v     = (const float*)d_in[7];
  const float* lin   = (const float*)d_in[8];

  const int N = in_sizes[1];   // node_weights
  const int E = in_sizes[2];   // edge_weights
  const int F = in_sizes[3];   // face_weights

  float* nh  = (float*)d_ws;   // N*32 floats (6.4 MB) of scratch
  float* out = (float*)d_out;

  // K0: zero output (harness poisons d_out)
  wdect_zero<<<(out_size + 255) / 256, 256, 0, stream>>>(out, out_size);

  // K1: node heights via f32 WMMA (one wave per 16-node tile)
  const int tiles = (N + 15) / 16;
  const int wpb = 8;                       // 256 threads = 8 waves
  wdect_heights<<<(tiles + wpb - 1) / wpb, 256, 0, stream>>>(x, nw, v, nh, N);

  // K2: ECC accumulation, 256KB dynamic LDS per block (1 block / WGP)
  wdect_ecc<<<384, 1024, ACC_ELEMS * sizeof(float), stream>>>(
      nh, batch, eidx, ew, fidx, fw, lin, out, N, E, F);
}